// Reader_62818191672010
// MI455X (gfx1250) — compile-verified
//
#include <hip/hip_runtime.h>
#include <math.h>

typedef __attribute__((ext_vector_type(2))) float v2f;
typedef __attribute__((ext_vector_type(8))) float v8f;
typedef __attribute__((ext_vector_type(4))) unsigned int v4u;
typedef __attribute__((ext_vector_type(4))) int v4i;
typedef __attribute__((ext_vector_type(8))) int v8i;

#define BB  64
#define NN  8192
#define MWW 128
#define HH  8
#define CC  1024
#define KK  3
#define HDD 128
#define EPSF 1e-8f

__device__ __forceinline__ float sigmoidf_(float x) { return 1.0f / (1.0f + expf(-x)); }
__device__ __forceinline__ float softplusf_(float x) { return (x > 20.0f) ? x : log1pf(expf(x)); }

// ---------------------------------------------------------------------------
// K0: zero the "headed" accumulator (atomically accumulated by ntm_read)
// ---------------------------------------------------------------------------
__global__ void ntm_zero(float* __restrict__ p, int n) {
  int i = blockIdx.x * blockDim.x + threadIdx.x;
  if (i < n) p[i] = 0.0f;
}

// ---------------------------------------------------------------------------
// K1: heads = control @ W_head + b_head   [64,1024] x [1024,1024]
// ---------------------------------------------------------------------------
__global__ void ntm_heads(const float* __restrict__ control,
                          const float* __restrict__ W_head,
                          const float* __restrict__ b_head,
                          float* __restrict__ heads) {
  __shared__ float csh[CC];
  const int b = blockIdx.x, t = threadIdx.x;
  csh[t] = control[b * CC + t];
  __syncthreads();
  float acc = b_head[t];
  for (int c = 0; c < CC; ++c)
    acc = fmaf(csh[c], W_head[(size_t)c * (HH * HDD) + t], acc);
  heads[(size_t)b * CC + t] = acc;
}

// ---------------------------------------------------------------------------
// K2: per-(b,h) params: key row (M-padded to 16 for WMMA), reset/beta/gate/
// gamma scalars, 3-tap shift softmax, key norm.
// params layout per (b,h): [reset_p, beta, gate, gamma, s0, s1, s2, keynorm]
// ---------------------------------------------------------------------------
__global__ void ntm_params(const float* __restrict__ heads,
                           const float* __restrict__ W_reset, const float* __restrict__ b_reset,
                           const float* __restrict__ W_key,   const float* __restrict__ b_key,
                           const float* __restrict__ W_beta,  const float* __restrict__ b_beta,
                           const float* __restrict__ W_gate,  const float* __restrict__ b_gate,
                           const float* __restrict__ W_shift, const float* __restrict__ b_shift,
                           const float* __restrict__ W_gamma, const float* __restrict__ b_gamma,
                           float* __restrict__ keyp, float* __restrict__ params) {
  const int b = blockIdx.x, h = blockIdx.y, t = threadIdx.x;
  if (h >= HH) {  // pad rows for WMMA M=16
    keyp[((size_t)b * 16 + h) * HDD + t] = 0.0f;
    return;
  }
  __shared__ float hsh[HDD];
  __shared__ float red[8 * HDD];
  hsh[t] = heads[((size_t)b * HH + h) * HDD + t];
  __syncthreads();

  float kv = b_key[h * MWW + t];
  for (int d = 0; d < HDD; ++d)
    kv = fmaf(hsh[d], W_key[((size_t)h * HDD + d) * MWW + t], kv);
  keyp[((size_t)b * 16 + h) * HDD + t] = kv;

  const float hv = hsh[t];
  red[0 * HDD + t] = hv * W_reset[h * HDD + t];
  red[1 * HDD + t] = hv * W_beta [h * HDD + t];
  red[2 * HDD + t] = hv * W_gate [h * HDD + t];
  red[3 * HDD + t] = hv * W_gamma[h * HDD + t];
  red[4 * HDD + t] = hv * W_shift[(h * HDD + t) * KK + 0];
  red[5 * HDD + t] = hv * W_shift[(h * HDD + t) * KK + 1];
  red[6 * HDD + t] = hv * W_shift[(h * HDD + t) * KK + 2];
  red[7 * HDD + t] = kv * kv;
  __syncthreads();
  for (int s = 64; s > 0; s >>= 1) {
    if (t < s)
      for (int q = 0; q < 8; ++q) red[q * HDD + t] += red[q * HDD + t + s];
    __syncthreads();
  }
  if (t == 0) {
    float* p = params + ((size_t)b * HH + h) * 8;
    p[0] = sigmoidf_(red[0 * HDD] + b_reset[h]);
    p[1] = softplusf_(red[1 * HDD] + b_beta[h]);
    p[2] = sigmoidf_(red[2 * HDD] + b_gate[h]);
    p[3] = 1.0f + softplusf_(red[3 * HDD] + b_gamma[h]);
    float e0 = red[4 * HDD] + b_shift[h * KK + 0];
    float e1 = red[5 * HDD] + b_shift[h * KK + 1];
    float e2 = red[6 * HDD] + b_shift[h * KK + 2];
    float mx = fmaxf(e0, fmaxf(e1, e2));
    float x0 = expf(e0 - mx), x1 = expf(e1 - mx), x2 = expf(e2 - mx);
    float si = 1.0f / (x0 + x1 + x2);
    p[4] = x0 * si; p[5] = x1 * si; p[6] = x2 * si;
    p[7] = sqrtf(red[7 * HDD]);
  }
}

// ---------------------------------------------------------------------------
// TDM: issue a tensor_load_to_lds of a [rows x 128] f32 2-D tile starting at
// global byte address `ga` into LDS byte offset `lds_off` (compacted rows).
// D# packing per CDNA5 ISA ch.8 (group0 128b, group1 256b, groups 2/3 zero).
// ---------------------------------------------------------------------------
__device__ __forceinline__ void tdm_load_tile_f32(unsigned long long ga,
                                                  unsigned int lds_off,
                                                  unsigned int rows) {
  v4u g0;
  g0[0] = 1u;                                          // count=1, user mode
  g0[1] = lds_off;                                     // lds_addr (bytes)
  g0[2] = (unsigned int)(ga & 0xffffffffull);          // global_addr[31:0]
  g0[3] = (unsigned int)((ga >> 32) & 0x1ffffffull)    // global_addr[56:32]
          | (2u << 30);                                // type=2 ("image")
  v8i g1;
  g1[0] = (int)(2u << 16);          // wg_mask=0, data_size=2 (4 bytes)
  g1[1] = (int)(128u << 16);        // tensor_dim0 = 128 (low 16 @ [63:48])
  g1[2] = (int)(rows << 16);        // dim0 hi=0, tensor_dim1 = rows
  g1[3] = (int)(128u << 16);        // dim1 hi=0, tile_dim0 = 128
  g1[4] = (int)rows;                // tile_dim1 = rows, tile_dim2 = 0
  g1[5] = 128;                      // tensor_dim0_stride = 128 elements
  g1[6] = 0;                        // stride hi, dim1_stride lo
  g1[7] = 0;                        // dim1_stride hi
  v4i g2 = {0, 0, 0, 0};
  v4i g3 = {0, 0, 0, 0};
#if defined(__clang_major__) && (__clang_major__ >= 23)
  v8i g4 = {0, 0, 0, 0, 0, 0, 0, 0};
  __builtin_amdgcn_tensor_load_to_lds(g0, g1, g2, g3, g4, 0);
#else
  __builtin_amdgcn_tensor_load_to_lds(g0, g1, g2, g3, 0);
#endif
}

// ---------------------------------------------------------------------------
// K3: sim[b,h,n] = beta_h * <key_h, mem_n> / (|key_h| * |mem_n| + eps)
// TDM stages the 64x128 f32 memory tile (32KB) into LDS; row norms and WMMA
// B-fragments are served from LDS. WMMA f32 16x16x4, M=16 (8 heads padded),
// K=128, one 16-slot N-tile per wave. grid (B, N/64), block 128 (4 waves).
// ---------------------------------------------------------------------------
__global__ void ntm_sim(const float* __restrict__ mem,
                        const float* __restrict__ keyp,
                        const float* __restrict__ params,
                        float* __restrict__ cbuf) {
  const int b  = blockIdx.x;
  const int n0 = blockIdx.y * 64;
  const int t  = threadIdx.x;
  __shared__ __align__(16) float tile[64 * MWW];   // 32KB DMA destination
  __shared__ float mn[64];
  __shared__ float pr[128];
  __shared__ float pl[HH * 8];

  if (t < 32) {  // one wave owns the TDM descriptor (TENSORcnt is per-wave)
    unsigned long long ga =
        (unsigned long long)(size_t)(mem + ((size_t)b * NN + n0) * MWW);
    tdm_load_tile_f32(ga, (unsigned int)(size_t)&tile[0], 64u);
    __builtin_amdgcn_s_wait_tensorcnt(0);
  }
  if (t < HH * 8) pl[t] = params[(size_t)(b * HH) * 8 + t];
  __syncthreads();

  // memory-row L2 norms for the 64 slots of this tile (2 threads per row)
  {
    const int row = t >> 1, half = t & 1;
    const float* mrow = &tile[row * MWW + half * 64];
    float s = 0.0f;
    for (int i = 0; i < 64; ++i) { float v = mrow[i]; s = fmaf(v, v, s); }
    pr[t] = s;
  }
  __syncthreads();
  if (t < 64) mn[t] = sqrtf(pr[2 * t] + pr[2 * t + 1]);
  __syncthreads();

  const int wave = t >> 5, lane = t & 31;
  const int m    = lane & 15;          // A: lane -> M row
  const int koff = (lane >> 4) * 2;    // lane half -> K pair
  const int col  = lane & 15;          // B: lane -> N column
  const float* arow = keyp + ((size_t)b * 16 + m) * HDD;
  const float* brow = &tile[(wave * 16 + col) * MWW];

  v8f cc = {0.f, 0.f, 0.f, 0.f, 0.f, 0.f, 0.f, 0.f};
#pragma unroll 4
  for (int k0 = 0; k0 < HDD; k0 += 4) {
    v2f av = *(const v2f*)(arow + k0 + koff);
    v2f bv = *(const v2f*)(brow + k0 + koff);
    cc = __builtin_amdgcn_wmma_f32_16x16x4_f32(false, av, false, bv, (short)0, cc, false, false);
  }

  // C/D layout: lanes 0-15 hold M=0..7 (our real heads) in elements r=0..7
  if (lane < 16) {
    const int n = n0 + wave * 16 + lane;
    const float mnv = mn[wave * 16 + lane];
#pragma unroll
    for (int r = 0; r < HH; ++r) {
      const float beta = pl[r * 8 + 1], kn = pl[r * 8 + 7];
      cbuf[((size_t)(b * HH + r)) * NN + n] = beta * cc[r] / (kn * mnv + EPSF);
    }
  }
}

// ---------------------------------------------------------------------------
// K4: per-(b,h) softmax over N, then gate/reset interpolation (in-place)
// ---------------------------------------------------------------------------
__global__ void ntm_softmax_wg(const float* __restrict__ params,
                               const float* __restrict__ prevw,
                               const float* __restrict__ defs,
                               float* __restrict__ cbuf) {
  const int bh = blockIdx.x, t = threadIdx.x;
  __shared__ float red[256];
  float* cb = cbuf + (size_t)bh * NN;

  float m = -INFINITY;
  for (int i = t; i < NN; i += 256) m = fmaxf(m, cb[i]);
  red[t] = m; __syncthreads();
  for (int s = 128; s > 0; s >>= 1) { if (t < s) red[t] = fmaxf(red[t], red[t + s]); __syncthreads(); }
  m = red[0]; __syncthreads();

  float se = 0.0f;
  for (int i = t; i < NN; i += 256) se += expf(cb[i] - m);
  red[t] = se; __syncthreads();
  for (int s = 128; s > 0; s >>= 1) { if (t < s) red[t] += red[t + s]; __syncthreads(); }
  const float inv = 1.0f / red[0];

  const float reset = params[bh * 8 + 0];
  const float gate  = params[bh * 8 + 2];
  const float* pw = prevw + (size_t)bh * NN;
  const float* dw = defs  + (size_t)bh * NN;
  for (int i = t; i < NN; i += 256) {
    float content = expf(cb[i] - m) * inv;
    float wsel = pw[i] * (1.0f - reset) + dw[i] * reset;
    cb[i] = gate * content + (1.0f - gate) * wsel;
  }
}

// ---------------------------------------------------------------------------
// K5: circular 3-tap shift conv + gamma sharpening + renormalize -> new_w
// Whole 8192-element row lives in LDS (32KB of the 320KB/WGP budget).
// ---------------------------------------------------------------------------
__global__ void ntm_conv_pow(const float* __restrict__ params,
                             const float* __restrict__ cbuf,
                             float* __restrict__ neww) {
  const int bh = blockIdx.x, t = threadIdx.x;  // 256 threads
  __shared__ float wg[NN];
  __shared__ float red[256];
  const float* cb = cbuf + (size_t)bh * NN;
  for (int i = t; i < NN; i += 256) wg[i] = cb[i];
  __syncthreads();

  const float s0 = params[bh * 8 + 4], s1 = params[bh * 8 + 5];
  const float s2 = params[bh * 8 + 6], gamma = params[bh * 8 + 3];
  float pv[32];
  float loc = 0.0f;
  const int base = t * 32;
#pragma unroll 4
  for (int j = 0; j < 32; ++j) {
    const int n = base + j;
    // jnp.roll(w, s)[n] = w[(n-s) mod N]; taps at s = -1, 0, +1
    float ws = s0 * wg[(n + 1) & (NN - 1)] + s1 * wg[n] + s2 * wg[(n - 1) & (NN - 1)];
    float p = powf(ws + EPSF, gamma);
    pv[j] = p; loc += p;
  }
  red[t] = loc; __syncthreads();
  for (int s = 128; s > 0; s >>= 1) { if (t < s) red[t] += red[t + s]; __syncthreads(); }
  const float inv = 1.0f / red[0];
  float* nw = neww + (size_t)bh * NN;
  for (int j = 0; j < 32; ++j) nw[base + j] = pv[j] * inv;
}

// ---------------------------------------------------------------------------
// K6: headed[b,h,:] = new_w[b,h,:] @ memory[b]   via WMMA f32 16x16x4
// grid (B, MW/16, 16 K-chunks), 1 wave per block, f32 atomics into headed.
// ---------------------------------------------------------------------------
__global__ void ntm_read(const float* __restrict__ mem,
                         const float* __restrict__ neww,
                         float* __restrict__ headed) {
  const int b      = blockIdx.x;
  const int w0     = blockIdx.y * 16;
  const int k0base = blockIdx.z * 512;
  const int lane = threadIdx.x;
  const int m = lane & 15, half = lane >> 4, col = lane & 15;
  const bool am = (m < HH);
  const float* arow = neww + ((size_t)(b * HH + (am ? m : 0))) * NN;
  const float* mb = mem + (size_t)b * NN * MWW + w0 + col;

  v8f cc = {0.f, 0.f, 0.f, 0.f, 0.f, 0.f, 0.f, 0.f};
  for (int k0 = k0base; k0 < k0base + 512; k0 += 4) {
    __builtin_prefetch(mb + (size_t)(k0 + 32) * MWW, 0, 1);
    v2f av;
    if (am) av = *(const v2f*)(arow + k0 + 2 * half);
    else    { av.x = 0.0f; av.y = 0.0f; }
    v2f bv;
    bv.x = mb[(size_t)(k0 + 2 * half) * MWW];
    bv.y = mb[(size_t)(k0 + 2 * half + 1) * MWW];
    cc = __builtin_amdgcn_wmma_f32_16x16x4_f32(false, av, false, bv, (short)0, cc, false, false);
  }
  if (lane < 16) {
#pragma unroll
    for (int r = 0; r < HH; ++r)
      atomicAdd(&headed[(size_t)(b * HH + r) * MWW + w0 + lane], cc[r]);
  }
}

// ---------------------------------------------------------------------------
// K7: out[b,w] = sum_h merge_w[h] * headed[b,h,w]
// ---------------------------------------------------------------------------
__global__ void ntm_merge(const float* __restrict__ headed,
                          const float* __restrict__ merge_w,
                          float* __restrict__ out) {
  const int b = blockIdx.x, w = threadIdx.x;
  float acc = 0.0f;
#pragma unroll
  for (int h = 0; h < HH; ++h)
    acc = fmaf(merge_w[h], headed[(size_t)(b * HH + h) * MWW + w], acc);
  out[b * MWW + w] = acc;
}

// ---------------------------------------------------------------------------
extern "C" void kernel_launch(void* const* d_in, const int* in_sizes, int n_in,
                              void* d_out, int out_size, void* d_ws, size_t ws_size,
                              hipStream_t stream) {
  (void)in_sizes; (void)n_in; (void)out_size; (void)ws_size;
  const float* control = (const float*)d_in[0];
  const float* memory  = (const float*)d_in[1];
  const float* prevw   = (const float*)d_in[2];
  const float* defs    = (const float*)d_in[3];
  const float* W_head  = (const float*)d_in[4];
  const float* b_head  = (const float*)d_in[5];
  const float* W_reset = (const float*)d_in[6];
  const float* b_reset = (const float*)d_in[7];
  const float* W_key   = (const float*)d_in[8];
  const float* b_key   = (const float*)d_in[9];
  const float* W_beta  = (const float*)d_in[10];
  const float* b_beta  = (const float*)d_in[11];
  const float* W_gate  = (const float*)d_in[12];
  const float* b_gate  = (const float*)d_in[13];
  const float* W_shift = (const float*)d_in[14];
  const float* b_shift = (const float*)d_in[15];
  const float* W_gamma = (const float*)d_in[16];
  const float* b_gamma = (const float*)d_in[17];
  const float* merge_w = (const float*)d_in[18];

  float* ws     = (float*)d_ws;
  float* heads  = ws;                 // 65536 f32
  float* keyp   = ws + 65536;         // 131072 f32  [B,16,HD] padded
  float* params = ws + 196608;        // 4096 f32    [B,H,8]
  float* headed = ws + 200704;        // 65536 f32   [B,H,MW]
  float* cbuf   = ws + 266240;        // 4194304 f32 [B,H,N]

  float* out  = (float*)d_out;        // [B,MW] then [B,H,N]
  float* neww = out + BB * MWW;

  ntm_zero<<<(BB * HH * MWW + 255) / 256, 256, 0, stream>>>(headed, BB * HH * MWW);
  ntm_heads<<<BB, CC, 0, stream>>>(control, W_head, b_head, heads);
  ntm_params<<<dim3(BB, 16), 128, 0, stream>>>(heads,
      W_reset, b_reset, W_key, b_key, W_beta, b_beta, W_gate, b_gate,
      W_shift, b_shift, W_gamma, b_gamma, keyp, params);
  ntm_sim<<<dim3(BB, NN / 64), 128, 0, stream>>>(memory, keyp, params, cbuf);
  ntm_softmax_wg<<<BB * HH, 256, 0, stream>>>(params, prevw, defs, cbuf);
  ntm_conv_pow<<<BB * HH, 256, 0, stream>>>(params, cbuf, neww);
  ntm_read<<<dim3(BB, MWW / 16, 16), 32, 0, stream>>>(memory, neww, headed);
  ntm_merge<<<BB, MWW, 0, stream>>>(headed, merge_w, out);
}